// CausalSelfAttention_56152402428101
// MI455X (gfx1250) — compile-verified
//
#include <hip/hip_runtime.h>

typedef __attribute__((ext_vector_type(2))) float v2f;
typedef __attribute__((ext_vector_type(8))) float v8f;

#define T_SEQ 4096
#define C_DIM 1024
#define NH    16
#define HD    64
#define HALF  32    // d/2 for RoPE

__device__ __forceinline__ v2f ld2(const float* p) { return *(const v2f*)p; }

__device__ __forceinline__ v8f wmma_f32(v2f a, v2f b, v8f c) {
    return __builtin_amdgcn_wmma_f32_16x16x4_f32(false, a, false, b, (short)0, c, false, false);
}

// ---------------------------------------------------------------------------
// Kernel 1: fused QKV projection (y = x @ W^T) + RoPE epilogue for Q,K.
// One wave per block; each wave computes a 16x64 tile (one head's columns),
// writes head-major [h][T][64] so attention reads contiguous rows.
// grid = 3 * 16 * 256 blocks of 32 threads.
// ---------------------------------------------------------------------------
__global__ __launch_bounds__(32) void qkv_rope_kernel(
    const float* __restrict__ x, const float* __restrict__ fc, const float* __restrict__ fs,
    const float* __restrict__ Wq, const float* __restrict__ Wk, const float* __restrict__ Wv,
    float* __restrict__ q_ws, float* __restrict__ k_ws, float* __restrict__ v_ws)
{
    __shared__ float tile[16 * 64];
    const int lane = threadIdx.x;
    int bid = blockIdx.x;
    const int mat = bid / (NH * 256);       // 0=Q 1=K 2=V
    bid %= (NH * 256);
    const int ct = bid / 256;               // column tile == head
    const int rt = bid % 256;               // row tile (16 rows)
    const int t0 = rt * 16;
    const int o0 = ct * 64;

    const float* W   = (mat == 0) ? Wq : (mat == 1) ? Wk : Wv;
    float*       dst = (mat == 0) ? q_ws : (mat == 1) ? k_ws : v_ws;

    const int arow = lane & 15;             // row/col within tile for this lane
    const int koff = (lane >> 4) * 2;       // K-half selector (ISA A/B layout)

    v8f acc0 = {}, acc1 = {}, acc2 = {}, acc3 = {};
    const float* xrow = x + (size_t)(t0 + arow) * C_DIM + koff;
    const float* w0   = W + (size_t)(o0 + arow) * C_DIM + koff;

    for (int c = 0; c < C_DIM; c += 4) {
        v2f a  = ld2(xrow + c);
        v2f b0 = ld2(w0 + c);
        v2f b1 = ld2(w0 + 16 * C_DIM + c);
        v2f b2 = ld2(w0 + 32 * C_DIM + c);
        v2f b3 = ld2(w0 + 48 * C_DIM + c);
        acc0 = wmma_f32(a, b0, acc0);
        acc1 = wmma_f32(a, b1, acc1);
        acc2 = wmma_f32(a, b2, acc2);
        acc3 = wmma_f32(a, b3, acc3);
    }

    // C/D layout: VGPR r -> row r (lanes 0-15) or row r+8 (lanes 16-31)
    const int rbase = (lane >> 4) << 3;
#pragma unroll
    for (int r = 0; r < 8; ++r) {
        const int row = rbase + r;
        tile[row * 64 +      arow] = acc0[r];
        tile[row * 64 + 16 + arow] = acc1[r];
        tile[row * 64 + 32 + arow] = acc2[r];
        tile[row * 64 + 48 + arow] = acc3[r];
    }
    __syncthreads();

    const bool dorope = (mat < 2);
    for (int idx = lane; idx < 16 * HALF; idx += 32) {
        const int row = idx >> 5;           // 0..15
        const int i   = idx & 31;           // pair index 0..31
        float s0 = tile[row * 64 + 2 * i];
        float s1 = tile[row * 64 + 2 * i + 1];
        float o0v = s0, o1v = s1;
        if (dorope) {
            const float cv = fc[(size_t)(t0 + row) * HALF + i];
            const float sv = fs[(size_t)(t0 + row) * HALF + i];
            o0v = s0 * cv - s1 * sv;
            o1v = s0 * sv + s1 * cv;
        }
        float* d = dst + ((size_t)ct * T_SEQ + (t0 + row)) * HD + 2 * i;
        d[0] = o0v;
        d[1] = o1v;
    }
}

// ---------------------------------------------------------------------------
// Kernel 2: causal flash attention, one wave per (head, 16-query tile).
// grid = (256, 16) blocks of 32 threads. Writes y in [T][C] layout.
// ---------------------------------------------------------------------------
__global__ __launch_bounds__(32) void attn_kernel(
    const float* __restrict__ q_ws, const float* __restrict__ k_ws,
    const float* __restrict__ v_ws, float* __restrict__ y_ws)
{
    __shared__ float sP[256];   // 16x16 score/prob tile
    __shared__ float srs[16];   // per-row broadcast scratch

    const int lane = threadIdx.x;
    const int qt   = blockIdx.x;
    const int head = blockIdx.y;
    const int t0   = qt * 16;

    const float* qh = q_ws + (size_t)head * T_SEQ * HD;
    const float* kh = k_ws + (size_t)head * T_SEQ * HD;
    const float* vh = v_ws + (size_t)head * T_SEQ * HD;

    const int arow  = lane & 15;
    const int koff  = (lane >> 4) * 2;
    const int rbase = (lane >> 4) << 3;

    // Q tile as 16 A-fragments (K-chunks of 4 along d)
    v2f qa[16];
#pragma unroll
    for (int t = 0; t < 16; ++t)
        qa[t] = ld2(qh + (size_t)(t0 + arow) * HD + 4 * t + koff);

    v8f o0 = {}, o1 = {}, o2 = {}, o3 = {};
    float m_run = -__builtin_inff();
    float l_run = 0.0f;
    const float scale = 0.125f;   // 1/sqrt(64)

    for (int j = 0; j <= qt; ++j) {
        const int k0 = j * 16;

        // S = Q @ K^T   (16x16 accumulator over d=64)
        v8f sacc = {};
#pragma unroll
        for (int t = 0; t < 16; ++t) {
            v2f kb = ld2(kh + (size_t)(k0 + arow) * HD + 4 * t + koff);
            sacc = wmma_f32(qa[t], kb, sacc);
        }

        // scale + causal mask, spill to LDS
#pragma unroll
        for (int r = 0; r < 8; ++r) {
            const int row = rbase + r;
            float s = sacc[r] * scale;
            if (j == qt && (k0 + arow) > (t0 + row)) s = -__builtin_inff();
            sP[row * 16 + arow] = s;
        }
        __syncthreads();

        // online softmax: lane r owns row r
        if (lane < 16) {
            float vals[16];
            float rowmax = -__builtin_inff();
#pragma unroll
            for (int c = 0; c < 16; ++c) {
                vals[c] = sP[lane * 16 + c];
                rowmax = fmaxf(rowmax, vals[c]);
            }
            const float m_new = fmaxf(m_run, rowmax);
            const float rs = __expf(m_run - m_new);
            float sum = 0.0f;
#pragma unroll
            for (int c = 0; c < 16; ++c) {
                const float p = __expf(vals[c] - m_new);
                sP[lane * 16 + c] = p;
                sum += p;
            }
            l_run = l_run * rs + sum;
            m_run = m_new;
            srs[lane] = rs;
        }
        __syncthreads();

        // rescale O accumulators by per-row factor
        float rsv[8];
#pragma unroll
        for (int r = 0; r < 8; ++r) rsv[r] = srs[rbase + r];
#pragma unroll
        for (int r = 0; r < 8; ++r) {
            o0[r] *= rsv[r]; o1[r] *= rsv[r]; o2[r] *= rsv[r]; o3[r] *= rsv[r];
        }

        // O += P @ V   (K-chunks of 4 over the 16 keys; 4 N-subtiles over d)
#pragma unroll
        for (int t = 0; t < 4; ++t) {
            v2f pa;
            pa[0] = sP[arow * 16 + 4 * t + koff];
            pa[1] = sP[arow * 16 + 4 * t + koff + 1];
            const float* vb0 = vh + (size_t)(k0 + 4 * t + koff) * HD;
            const float* vb1 = vb0 + HD;
#pragma unroll
            for (int nn = 0; nn < 4; ++nn) {
                v2f b;
                b[0] = vb0[16 * nn + arow];
                b[1] = vb1[16 * nn + arow];
                if      (nn == 0) o0 = wmma_f32(pa, b, o0);
                else if (nn == 1) o1 = wmma_f32(pa, b, o1);
                else if (nn == 2) o2 = wmma_f32(pa, b, o2);
                else              o3 = wmma_f32(pa, b, o3);
            }
        }
        __syncthreads();
    }

    // final 1/l normalization, write y[t][head*64 + dd]
    if (lane < 16) srs[lane] = 1.0f / l_run;
    __syncthreads();
    float inv[8];
#pragma unroll
    for (int r = 0; r < 8; ++r) inv[r] = srs[rbase + r];
#pragma unroll
    for (int r = 0; r < 8; ++r) {
        const int row = rbase + r;
        float* y = y_ws + (size_t)(t0 + row) * C_DIM + head * HD;
        y[      arow] = o0[r] * inv[r];
        y[16 +  arow] = o1[r] * inv[r];
        y[32 +  arow] = o2[r] * inv[r];
        y[48 +  arow] = o3[r] * inv[r];
    }
}

// ---------------------------------------------------------------------------
// Kernel 3: output projection out = y @ Wo^T.  grid = 16*256 blocks of 32.
// ---------------------------------------------------------------------------
__global__ __launch_bounds__(32) void oproj_kernel(
    const float* __restrict__ y_ws, const float* __restrict__ Wo,
    float* __restrict__ out)
{
    const int lane = threadIdx.x;
    const int ct = blockIdx.x / 256;
    const int rt = blockIdx.x % 256;
    const int t0 = rt * 16;
    const int o0 = ct * 64;

    const int arow = lane & 15;
    const int koff = (lane >> 4) * 2;

    v8f acc0 = {}, acc1 = {}, acc2 = {}, acc3 = {};
    const float* yrow = y_ws + (size_t)(t0 + arow) * C_DIM + koff;
    const float* w0   = Wo   + (size_t)(o0 + arow) * C_DIM + koff;

    for (int c = 0; c < C_DIM; c += 4) {
        v2f a  = ld2(yrow + c);
        v2f b0 = ld2(w0 + c);
        v2f b1 = ld2(w0 + 16 * C_DIM + c);
        v2f b2 = ld2(w0 + 32 * C_DIM + c);
        v2f b3 = ld2(w0 + 48 * C_DIM + c);
        acc0 = wmma_f32(a, b0, acc0);
        acc1 = wmma_f32(a, b1, acc1);
        acc2 = wmma_f32(a, b2, acc2);
        acc3 = wmma_f32(a, b3, acc3);
    }

    const int rbase = (lane >> 4) << 3;
#pragma unroll
    for (int r = 0; r < 8; ++r) {
        const int row = rbase + r;
        float* d = out + (size_t)(t0 + row) * C_DIM + o0;
        d[      arow] = acc0[r];
        d[16 +  arow] = acc1[r];
        d[32 +  arow] = acc2[r];
        d[48 +  arow] = acc3[r];
    }
}

// ---------------------------------------------------------------------------
extern "C" void kernel_launch(void* const* d_in, const int* in_sizes, int n_in,
                              void* d_out, int out_size, void* d_ws, size_t ws_size,
                              hipStream_t stream) {
    const float* x  = (const float*)d_in[0];
    const float* fc = (const float*)d_in[1];
    const float* fs = (const float*)d_in[2];
    const float* Wq = (const float*)d_in[3];
    const float* Wk = (const float*)d_in[4];
    const float* Wv = (const float*)d_in[5];
    const float* Wo = (const float*)d_in[6];
    float* out = (float*)d_out;

    float* ws   = (float*)d_ws;
    float* q_ws = ws;                        // [16][4096][64]
    float* k_ws = ws + 4194304;              // [16][4096][64]
    float* v_ws = ws + 8388608;              // [16][4096][64]
    float* y_ws = ws + 12582912;             // [4096][1024]

    qkv_rope_kernel<<<dim3(3 * NH * 256), dim3(32), 0, stream>>>(
        x, fc, fs, Wq, Wk, Wv, q_ws, k_ws, v_ws);
    attn_kernel<<<dim3(256, NH), dim3(32), 0, stream>>>(q_ws, k_ws, v_ws, y_ws);
    oproj_kernel<<<dim3(NH * 256), dim3(32), 0, stream>>>(y_ws, Wo, out);
}